// AstEmbed_11381663334406
// MI455X (gfx1250) — compile-verified
//
#include <hip/hip_runtime.h>
#include <cstdint>

typedef float v2f __attribute__((ext_vector_type(2)));
typedef float v8f __attribute__((ext_vector_type(8)));

#define DIM 128
#define NEG_SLOPE 0.2f
#define KEY_NEG_INF 0x007FFFFFu   // sortable key of -inf

__device__ __forceinline__ unsigned fkey(float f) {
    unsigned b = __float_as_uint(f);
    return (b & 0x80000000u) ? ~b : (b | 0x80000000u);
}
__device__ __forceinline__ float funkey(unsigned k) {
    unsigned b = (k & 0x80000000u) ? (k & 0x7FFFFFFFu) : ~k;
    return __uint_as_float(b);
}
__device__ __forceinline__ float lrelu(float x) {
    return x > 0.0f ? x : NEG_SLOPE * x;
}

// ---------------------------------------------------------------------------
// xl = h @ Wl + bl ; xr = h @ Wr + br   via V_WMMA_F32_16X16X4_F32
// grid.x = N/16 row tiles; 8 waves/block, wave w owns N-columns [16w,16w+16)
// ---------------------------------------------------------------------------
__global__ __launch_bounds__(256) void gemm_xlxr(
    const float* __restrict__ h,
    const float* __restrict__ Wl, const float* __restrict__ bl,
    const float* __restrict__ Wr, const float* __restrict__ br,
    float* __restrict__ xl, float* __restrict__ xr, int N)
{
    const int m0   = blockIdx.x * 16;
    const int wave = threadIdx.x >> 5;
    const int n0   = wave * 16;
    const int lane = threadIdx.x & 31;
    const int half = lane >> 4;      // 0: K pair {0,1}; 1: K pair {2,3}
    const int l16  = lane & 15;

    int arowi = m0 + l16; if (arowi >= N) arowi = N - 1;
    const float* arow = h + (size_t)arowi * DIM + 2 * half;

    v8f accl = {};
    v8f accr = {};
    for (int k = 0; k < DIM; k += 4) {
        // A fragment: row (m0+l16), cols k+2*half, k+2*half+1  (8B aligned)
        v2f a = *(const v2f*)(arow + k);
        // B fragments: W[k+2*half + {0,1}][n0+l16]
        const int r0 = (k + 2 * half) * DIM + n0 + l16;
        v2f b0, b1;
        b0.x = Wl[r0]; b0.y = Wl[r0 + DIM];
        b1.x = Wr[r0]; b1.y = Wr[r0 + DIM];
        accl = __builtin_amdgcn_wmma_f32_16x16x4_f32(false, a, false, b0,
                                                     (short)0, accl, false, false);
        accr = __builtin_amdgcn_wmma_f32_16x16x4_f32(false, a, false, b1,
                                                     (short)0, accr, false, false);
    }
    const float bLv = bl[n0 + l16];
    const float bRv = br[n0 + l16];
    const int mb = m0 + 8 * half;    // C/D: VGPR i = row mb+i, col n0+l16
#pragma unroll
    for (int i = 0; i < 8; ++i) {
        const int row = mb + i;
        if (row < N) {
            const size_t off = (size_t)row * DIM + n0 + l16;
            xl[off] = accl[i] + bLv;
            xr[off] = accr[i] + bRv;
        }
    }
}

// ---------------------------------------------------------------------------
__global__ __launch_bounds__(256) void init_accum(
    float* __restrict__ outacc, float* __restrict__ z,
    unsigned* __restrict__ mkey, int N)
{
    const int idx = blockIdx.x * blockDim.x + threadIdx.x;
    if (idx < N * DIM) outacc[idx] = 0.0f;
    if (idx < N) { z[idx] = 0.0f; mkey[idx] = KEY_NEG_INF; }
}

// ---------------------------------------------------------------------------
// One wave32 per edge: e = att . leaky_relu(xl[src] + xr[dst]); seg-max(dst)
// Edges [0,E): from edge_index (int64).  Edges [E,E+N): self loops.
// ---------------------------------------------------------------------------
__global__ __launch_bounds__(256) void edge_score(
    const float* __restrict__ xl, const float* __restrict__ xr,
    const long long* __restrict__ ei, int E, int Etot,
    const float* __restrict__ att,
    float* __restrict__ ebuf, unsigned* __restrict__ mkey)
{
    const int w = (int)((blockIdx.x * (unsigned)blockDim.x + threadIdx.x) >> 5);
    if (w >= Etot) return;
    const int lane = threadIdx.x & 31;
    long long s, d;
    if (w < E) { s = ei[w]; d = ei[(size_t)E + w]; }
    else       { s = d = (long long)(w - E); }

    const float4 vl = ((const float4*)(xl + (size_t)s * DIM))[lane];
    const float4 vr = ((const float4*)(xr + (size_t)d * DIM))[lane];
    const float4 av = ((const float4*)att)[lane];

    float part = lrelu(vl.x + vr.x) * av.x
               + lrelu(vl.y + vr.y) * av.y
               + lrelu(vl.z + vr.z) * av.z
               + lrelu(vl.w + vr.w) * av.w;
#pragma unroll
    for (int off = 16; off > 0; off >>= 1)
        part += __shfl_xor(part, off, 32);

    if (lane == 0) {
        ebuf[w] = part;
        atomicMax(&mkey[(size_t)d], fkey(part));
    }
}

// ---------------------------------------------------------------------------
// p = exp(e - m[dst]); z[dst] += p; out[dst] += p * xl[src]
// (normalization by z deferred to node_finalize -> one fewer edge pass)
// ---------------------------------------------------------------------------
__global__ __launch_bounds__(256) void edge_accum(
    const float* __restrict__ xl,
    const long long* __restrict__ ei, int E, int Etot,
    const float* __restrict__ ebuf, const unsigned* __restrict__ mkey,
    float* __restrict__ z, float* __restrict__ outacc)
{
    const int w = (int)((blockIdx.x * (unsigned)blockDim.x + threadIdx.x) >> 5);
    if (w >= Etot) return;
    const int lane = threadIdx.x & 31;
    long long s, d;
    if (w < E) { s = ei[w]; d = ei[(size_t)E + w]; }
    else       { s = d = (long long)(w - E); }

    const float m = funkey(mkey[(size_t)d]);
    const float p = __expf(ebuf[w] - m);
    if (lane == 0) unsafeAtomicAdd(&z[(size_t)d], p);

    const float4 vl = ((const float4*)(xl + (size_t)s * DIM))[lane];
    float* o = outacc + (size_t)d * DIM + lane * 4;
    unsafeAtomicAdd(o + 0, p * vl.x);
    unsafeAtomicAdd(o + 1, p * vl.y);
    unsafeAtomicAdd(o + 2, p * vl.z);
    unsafeAtomicAdd(o + 3, p * vl.w);
}

// ---------------------------------------------------------------------------
__global__ __launch_bounds__(256) void node_finalize(
    const float* __restrict__ outacc, const float* __restrict__ z,
    const float* __restrict__ bias, float* __restrict__ h,
    int N, int do_relu)
{
    const int idx = blockIdx.x * blockDim.x + threadIdx.x;
    if (idx >= N * DIM) return;
    const int n = idx >> 7;
    const int c = idx & (DIM - 1);
    float v = outacc[idx] / fmaxf(z[n], 1e-16f) + bias[c];
    if (do_relu) v = fmaxf(v, 0.0f);
    h[idx] = v;
}

// ---------------------------------------------------------------------------
__global__ void zero_out128(float* __restrict__ o) { o[threadIdx.x] = 0.0f; }

__global__ __launch_bounds__(128) void mean_reduce(
    const float* __restrict__ h, float* __restrict__ o, int N)
{
    const int c = threadIdx.x;     // column 0..127
    float s = 0.0f;
    for (int n = blockIdx.x; n < N; n += gridDim.x)
        s += h[(size_t)n * DIM + c];
    unsafeAtomicAdd(&o[c], s / (float)N);
}

// ---------------------------------------------------------------------------
extern "C" void kernel_launch(void* const* d_in, const int* in_sizes, int n_in,
                              void* d_out, int out_size, void* d_ws, size_t ws_size,
                              hipStream_t stream)
{
    const float*     x    = (const float*)d_in[0];
    const long long* ei   = (const long long*)d_in[1];   // int64 per reference
    const float*     Wl   = (const float*)d_in[2];
    const float*     bl   = (const float*)d_in[3];
    const float*     Wr   = (const float*)d_in[4];
    const float*     br   = (const float*)d_in[5];
    const float*     att  = (const float*)d_in[6];
    const float*     bias = (const float*)d_in[7];

    const int N      = in_sizes[0] / DIM;      // 50000
    const int E      = in_sizes[1] / 2;        // 1,600,000
    const int Etot   = E + N;
    const int LAYERS = in_sizes[3] / DIM;      // 5

    // ---- workspace layout -------------------------------------------------
    char* ws = (char*)d_ws;
    auto take = [&](size_t bytes) {
        char* p = ws;
        ws += (bytes + 255) & ~(size_t)255;
        return p;
    };
    const size_t nodef = (size_t)N * DIM * sizeof(float);
    float*    hbuf   = (float*)take(nodef);
    float*    xl     = (float*)take(nodef);
    float*    xr     = (float*)take(nodef);
    float*    outac  = (float*)take(nodef);
    float*    ebuf   = (float*)take((size_t)Etot * sizeof(float));
    float*    z      = (float*)take((size_t)N * sizeof(float));
    unsigned* mk     = (unsigned*)take((size_t)N * sizeof(unsigned));
    (void)ws_size;

    const int gemm_blocks = (N + 15) / 16;
    const int node_blocks = (N * DIM + 255) / 256;
    const int edge_blocks = (Etot + 7) / 8;      // 8 waves (edges) per block

    for (int l = 0; l < LAYERS; ++l) {
        const float* hin = (l == 0) ? x : hbuf;
        const float* Wl_l   = Wl   + (size_t)l * DIM * DIM;
        const float* Wr_l   = Wr   + (size_t)l * DIM * DIM;
        const float* bl_l   = bl   + (size_t)l * DIM;
        const float* br_l   = br   + (size_t)l * DIM;
        const float* att_l  = att  + (size_t)l * DIM;
        const float* bias_l = bias + (size_t)l * DIM;

        gemm_xlxr<<<gemm_blocks, 256, 0, stream>>>(hin, Wl_l, bl_l, Wr_l, br_l,
                                                   xl, xr, N);
        init_accum<<<node_blocks, 256, 0, stream>>>(outac, z, mk, N);
        edge_score<<<edge_blocks, 256, 0, stream>>>(xl, xr, ei, E, Etot,
                                                    att_l, ebuf, mk);
        edge_accum<<<edge_blocks, 256, 0, stream>>>(xl, ei, E, Etot,
                                                    ebuf, mk, z, outac);
        node_finalize<<<node_blocks, 256, 0, stream>>>(outac, z, bias_l, hbuf,
                                                       N, (l < LAYERS - 1) ? 1 : 0);
    }

    zero_out128<<<1, DIM, 0, stream>>>((float*)d_out);
    mean_reduce<<<512, DIM, 0, stream>>>(hbuf, (float*)d_out, N);
}